// CausalAttention_26431228739825
// MI455X (gfx1250) — compile-verified
//
#include <hip/hip_runtime.h>

typedef __attribute__((ext_vector_type(2))) float v2f;
typedef __attribute__((ext_vector_type(8))) float v8f;

#define B_    8
#define S_    4096
#define DIN_  512
#define DOUT_ 64

static __device__ __forceinline__ v8f wmma_f32(v2f a, v2f b, v8f c) {
  // D = A(16x4, f32) * B(4x16, f32) + C(16x16, f32)
  return __builtin_amdgcn_wmma_f32_16x16x4_f32(false, a, false, b, (short)0, c,
                                               false, false);
}

// ---------------------------------------------------------------------------
// Kernel 1: QKV projection.  One wave computes one 16x64 tile of Q, K or V
// (4 independent WMMA accumulator chains; A-fragment loaded once per k-step).
//   jobs = (M/16) * 3 matrices = 2048 * 3 = 6144 waves
// ---------------------------------------------------------------------------
__global__ __launch_bounds__(256) void qkv_proj_kernel(
    const float* __restrict__ x,  const float* __restrict__ wq,
    const float* __restrict__ wk, const float* __restrict__ wv,
    float* __restrict__ q, float* __restrict__ k, float* __restrict__ v)
{
  const int wave = (blockIdx.x * blockDim.x + threadIdx.x) >> 5;
  const int lane = threadIdx.x & 31;
  const int lo   = lane & 15;   // M (A) / N (B,C) index
  const int hi   = lane >> 4;   // selects K pair

  const int m_tile = wave / 3;
  const int mat    = wave % 3;        // 0=Q 1=K 2=V
  const int m0     = m_tile * 16;

  const float* W = (mat == 0) ? wq : (mat == 1) ? wk : wv;
  float*       O = (mat == 0) ? q  : (mat == 1) ? k  : v;

  v8f a0 = {}, a1 = {}, a2 = {}, a3 = {};
  const float* arow = x + (size_t)(m0 + lo) * DIN_ + hi * 2;
  #pragma unroll 4
  for (int kk = 0; kk < DIN_; kk += 4) {
    v2f a = *(const v2f*)(arow + kk);            // A[lo][kk+hi*2 .. +1]
    const float* w0 = W + (size_t)(kk + hi * 2) * DOUT_ + lo;
    const float* w1 = w0 + DOUT_;
    v2f b;
    b.x = w0[ 0]; b.y = w1[ 0]; a0 = wmma_f32(a, b, a0);
    b.x = w0[16]; b.y = w1[16]; a1 = wmma_f32(a, b, a1);
    b.x = w0[32]; b.y = w1[32]; a2 = wmma_f32(a, b, a2);
    b.x = w0[48]; b.y = w1[48]; a3 = wmma_f32(a, b, a3);
  }
  #pragma unroll
  for (int vv = 0; vv < 8; ++vv) {
    size_t ro = (size_t)(m0 + vv + 8 * hi) * DOUT_ + lo;
    O[ro +  0] = a0[vv];
    O[ro + 16] = a1[vv];
    O[ro + 32] = a2[vv];
    O[ro + 48] = a3[vv];
  }
}

// ---------------------------------------------------------------------------
// Kernel 2: causal flash attention.  One wave owns a 16-row q tile.
//   waves = B * S/16 = 8 * 256 = 2048
// ---------------------------------------------------------------------------
#define AW_ 4  // waves per block

__global__ __launch_bounds__(32 * AW_) void attn_kernel(
    const float* __restrict__ q, const float* __restrict__ k,
    const float* __restrict__ v, float* __restrict__ out)
{
  __shared__ float Plds[AW_][16][17];   // padded: gcd(17,32)=1 -> no conflicts
  const int wib  = threadIdx.x >> 5;
  const int lane = threadIdx.x & 31;
  const int lo   = lane & 15;
  const int hi   = lane >> 4;
  const int wave = blockIdx.x * AW_ + wib;

  const int b = wave >> 8;        // / (S/16)
  const int i = wave & 255;       // q tile index

  const float* qb = q + (size_t)b * S_ * DOUT_;
  const float* kb = k + (size_t)b * S_ * DOUT_;
  const float* vb = v + (size_t)b * S_ * DOUT_;

  // Q fragments: 16 K-steps of the 16x64 q tile, kept in registers.
  v2f qa[16];
  const float* qrow = qb + (size_t)(i * 16 + lo) * DOUT_ + hi * 2;
  #pragma unroll
  for (int t = 0; t < 16; ++t) qa[t] = *(const v2f*)(qrow + t * 4);

  v8f o0 = {}, o1 = {}, o2 = {}, o3 = {};
  float m[8], l[8];
  #pragma unroll
  for (int vv = 0; vv < 8; ++vv) { m[vv] = -INFINITY; l[vv] = 0.f; }

  const float SCALE = 0.125f * 1.44269504088896f;   // 1/sqrt(64) * log2(e)

  for (int j = 0; j <= i; ++j) {
    // ---- scores: S = q * K^T, two independent WMMA chains for ILP
    v8f sA = {}, sB = {};
    const float* krow = kb + (size_t)(j * 16 + lo) * DOUT_ + hi * 2;
    #pragma unroll
    for (int t = 0; t < 16; t += 2) {
      v2f bf0 = *(const v2f*)(krow + t * 4);
      v2f bf1 = *(const v2f*)(krow + t * 4 + 4);
      sA = wmma_f32(qa[t],     bf0, sA);
      sB = wmma_f32(qa[t + 1], bf1, sB);
    }
    v8f s;
    #pragma unroll
    for (int vv = 0; vv < 8; ++vv) s[vv] = (sA[vv] + sB[vv]) * SCALE;

    // ---- causal mask on the diagonal tile (row = vv+8*hi, col = lo)
    if (j == i) {
      #pragma unroll
      for (int vv = 0; vv < 8; ++vv)
        s[vv] = (lo > vv + 8 * hi) ? -INFINITY : s[vv];
    }

    // ---- online softmax per row (cols distributed across lanes lo)
    float alpha[8];
    #pragma unroll
    for (int vv = 0; vv < 8; ++vv) {
      float t = s[vv];
      t = fmaxf(t, __shfl_xor(t, 1, 32));
      t = fmaxf(t, __shfl_xor(t, 2, 32));
      t = fmaxf(t, __shfl_xor(t, 4, 32));
      t = fmaxf(t, __shfl_xor(t, 8, 32));
      float mn  = fmaxf(m[vv], t);
      alpha[vv] = exp2f(m[vv] - mn);
      m[vv]     = mn;
      float p   = exp2f(s[vv] - mn);
      float rs  = p;
      rs += __shfl_xor(rs, 1, 32);
      rs += __shfl_xor(rs, 2, 32);
      rs += __shfl_xor(rs, 4, 32);
      rs += __shfl_xor(rs, 8, 32);
      l[vv] = l[vv] * alpha[vv] + rs;
      Plds[wib][vv + 8 * hi][lo] = p;   // C-layout -> LDS (same-wave LDS is in-order)
    }

    // ---- rescale accumulators
    #pragma unroll
    for (int vv = 0; vv < 8; ++vv) {
      o0[vv] *= alpha[vv]; o1[vv] *= alpha[vv];
      o2[vv] *= alpha[vv]; o3[vv] *= alpha[vv];
    }

    // ---- O += P(16x16) * V(16x64): 4 K-steps x 4 N-tiles (4 indep. chains)
    #pragma unroll
    for (int kk = 0; kk < 4; ++kk) {
      v2f pa;
      pa.x = Plds[wib][lo][kk * 4 + hi * 2 + 0];   // A-layout read-back
      pa.y = Plds[wib][lo][kk * 4 + hi * 2 + 1];
      const float* vr0 = vb + (size_t)(j * 16 + kk * 4 + hi * 2) * DOUT_ + lo;
      const float* vr1 = vr0 + DOUT_;
      v2f bf;
      bf.x = vr0[ 0]; bf.y = vr1[ 0]; o0 = wmma_f32(pa, bf, o0);
      bf.x = vr0[16]; bf.y = vr1[16]; o1 = wmma_f32(pa, bf, o1);
      bf.x = vr0[32]; bf.y = vr1[32]; o2 = wmma_f32(pa, bf, o2);
      bf.x = vr0[48]; bf.y = vr1[48]; o3 = wmma_f32(pa, bf, o3);
    }
  }

  // ---- normalize and store
  float* ob = out + (size_t)b * S_ * DOUT_;
  #pragma unroll
  for (int vv = 0; vv < 8; ++vv) {
    float inv = 1.0f / l[vv];
    size_t ro = (size_t)(i * 16 + vv + 8 * hi) * DOUT_ + lo;
    ob[ro +  0] = o0[vv] * inv;
    ob[ro + 16] = o1[vv] * inv;
    ob[ro + 32] = o2[vv] * inv;
    ob[ro + 48] = o3[vv] * inv;
  }
}

// ---------------------------------------------------------------------------
extern "C" void kernel_launch(void* const* d_in, const int* in_sizes, int n_in,
                              void* d_out, int out_size, void* d_ws, size_t ws_size,
                              hipStream_t stream) {
  (void)in_sizes; (void)n_in; (void)out_size; (void)ws_size;
  const float* x  = (const float*)d_in[0];
  const float* wq = (const float*)d_in[1];
  const float* wk = (const float*)d_in[2];
  const float* wv = (const float*)d_in[3];

  float* q = (float*)d_ws;                       // B*S*64 floats = 8 MiB each
  float* k = q + (size_t)B_ * S_ * DOUT_;
  float* v = k + (size_t)B_ * S_ * DOUT_;
  float* o = (float*)d_out;

  // 6144 waves, 8 waves/block
  qkv_proj_kernel<<<768, 256, 0, stream>>>(x, wq, wk, wv, q, k, v);
  // 2048 waves, 4 waves/block
  attn_kernel<<<512, 32 * AW_, 0, stream>>>(q, k, v, o);
}